// LightGCN_47261820125825
// MI455X (gfx1250) — compile-verified
//
#include <hip/hip_runtime.h>

#define NUM_USERS 100000
#define NUM_ITEMS 50000
#define EMBED_DIM 64
#define N_NODES   (NUM_USERS + NUM_ITEMS)
#define N_EDGES   4800000
#define N_LAYERS  3
#define N_PAIRS   16384

typedef __attribute__((ext_vector_type(2))) float v2f;
typedef __attribute__((ext_vector_type(8))) float v8f;

// ---------------------------------------------------------------------------
// init: acc = h = concat(user_emb, item_emb); h_next = 0.  float4 vectorized.
// ---------------------------------------------------------------------------
__global__ void lgcn_init(const float* __restrict__ user_emb,
                          const float* __restrict__ item_emb,
                          float* __restrict__ acc,
                          float* __restrict__ h,
                          float* __restrict__ hnext) {
    const long total4 = (long)N_NODES * (EMBED_DIM / 4);
    long i = (long)blockIdx.x * blockDim.x + threadIdx.x;
    if (i >= total4) return;
    long node = i >> 4;          // 16 float4 per node
    int  c4   = (int)(i & 15);
    float4 v;
    if (node < NUM_USERS) {
        v = ((const float4*)(user_emb + node * EMBED_DIM))[c4];
    } else {
        v = ((const float4*)(item_emb + (node - NUM_USERS) * EMBED_DIM))[c4];
    }
    ((float4*)acc)[i]   = v;
    ((float4*)h)[i]     = v;
    ((float4*)hnext)[i] = make_float4(0.f, 0.f, 0.f, 0.f);
}

// ---------------------------------------------------------------------------
// spmm: EDGES_PER_WAVE (4) edges per wave32.  Each lane handles 2 dims via a float2 gather
// from h[col] (L2-resident, 38.4 MB table) and 2 global_atomic_add_f32 into
// hnext[row].  Edge metadata loads are wave-uniform -> single cacheline.
// ---------------------------------------------------------------------------
#define EDGES_PER_WAVE 4
__global__ void lgcn_spmm(const int*   __restrict__ rows,
                          const int*   __restrict__ cols,
                          const float* __restrict__ vals,
                          const float* __restrict__ h,
                          float*       __restrict__ hnext) {
    long tid  = (long)blockIdx.x * blockDim.x + threadIdx.x;
    long wave = tid >> 5;
    int  lane = (int)(tid & 31);
    long e0   = wave * EDGES_PER_WAVE;
#pragma unroll
    for (int k = 0; k < EDGES_PER_WAVE; ++k) {
        long e = e0 + k;
        if (e >= N_EDGES) break;
        int   r = rows[e];
        int   c = cols[e];
        float v = vals[e];
        const float2 x = *(const float2*)(h + (long)c * EMBED_DIM + lane * 2);
        float* dst = hnext + (long)r * EMBED_DIM + lane * 2;
        atomicAdd(dst + 0, v * x.x);
        atomicAdd(dst + 1, v * x.y);
    }
}

// ---------------------------------------------------------------------------
// fuse: acc += hnext;  zero the old h buffer so it can be the next target.
// ---------------------------------------------------------------------------
__global__ void lgcn_fuse(float* __restrict__ acc,
                          const float* __restrict__ hnext,
                          float* __restrict__ hold) {
    const long total4 = (long)N_NODES * (EMBED_DIM / 4);
    long i = (long)blockIdx.x * blockDim.x + threadIdx.x;
    if (i >= total4) return;
    float4 a  = ((float4*)acc)[i];
    float4 hn = ((const float4*)hnext)[i];
    a.x += hn.x; a.y += hn.y; a.z += hn.z; a.w += hn.w;
    ((float4*)acc)[i]  = a;
    ((float4*)hold)[i] = make_float4(0.f, 0.f, 0.f, 0.f);
}

// ---------------------------------------------------------------------------
// dot16: 16 user/item pairs per wave.  D = U(16x64) x V^T(64x16) via 16
// chained V_WMMA_F32_16X16X4_F32 (full f32 precision); diagonal of the 16x16
// tile holds the dot products.  A layout: lane m=L%16, VGPR j holds
// K = 2*(L/16)+j.  B layout (rows striped across lanes): lane n=L%16, VGPR j
// holds K = 2*(L/16)+j.  Both collapse to the same float2 row load.
// Final scale: (acc/4).(acc/4) -> 1/16.
// ---------------------------------------------------------------------------
__global__ void lgcn_dot16(const int* __restrict__ users,
                           const int* __restrict__ items,
                           const float* __restrict__ acc,
                           float* __restrict__ out) {
    __shared__ float tile[8][256];              // 8 waves * (32 lanes * 8 vgprs)
    const int lane = threadIdx.x & 31;
    const int warp = threadIdx.x >> 5;
    const int m    = lane & 15;                 // row of A == col of B
    const int half = lane >> 4;                 // K half select
    const int pairBase = blockIdx.x * 128 + warp * 16;

    const int u = users[pairBase + m];
    const int t = items[pairBase + m];
    const float* uRow = acc + (long)u * EMBED_DIM;
    const float* vRow = acc + ((long)NUM_USERS + t) * EMBED_DIM;

    v8f c = {};
#pragma unroll
    for (int kk = 0; kk < EMBED_DIM / 4; ++kk) {
        const int k = kk * 4 + half * 2;
        const float2 ua = *(const float2*)(uRow + k);
        const float2 vb = *(const float2*)(vRow + k);
        v2f a; a[0] = ua.x; a[1] = ua.y;
        v2f b; b[0] = vb.x; b[1] = vb.y;
        // 8 args: (neg_a, A, neg_b, B, c_mod, C, reuse_a, reuse_b)
        c = __builtin_amdgcn_wmma_f32_16x16x4_f32(false, a, false, b,
                                                  (short)0, c, false, false);
    }

    float* trow = &tile[warp][lane * 8];
#pragma unroll
    for (int j = 0; j < 8; ++j) trow[j] = c[j];
    __syncthreads();

    if (lane < 16) {
        // D[m][m]: m<8 -> (lane m, vgpr m); m>=8 -> (lane m+16, vgpr m-8)
        float d = (m < 8) ? tile[warp][m * 8 + m]
                          : tile[warp][(m + 16) * 8 + (m - 8)];
        out[pairBase + m] = d * 0.0625f;
    }
}

// ---------------------------------------------------------------------------
extern "C" void kernel_launch(void* const* d_in, const int* in_sizes, int n_in,
                              void* d_out, int out_size, void* d_ws, size_t ws_size,
                              hipStream_t stream) {
    const int*   users    = (const int*)  d_in[0];
    const int*   items    = (const int*)  d_in[1];
    const float* user_emb = (const float*)d_in[2];
    const float* item_emb = (const float*)d_in[3];
    const int*   adj_rows = (const int*)  d_in[4];
    const int*   adj_cols = (const int*)  d_in[5];
    const float* adj_vals = (const float*)d_in[6];
    float*       out      = (float*)d_out;

    const size_t NODE_FLOATS = (size_t)N_NODES * EMBED_DIM;   // 9.6M floats
    float* acc = (float*)d_ws;
    float* hA  = acc + NODE_FLOATS;    // current layer input
    float* hB  = hA + NODE_FLOATS;     // scatter target (zeroed)

    const long total4    = (long)N_NODES * (EMBED_DIM / 4);
    const int  eltBlocks = (int)((total4 + 255) / 256);

    lgcn_init<<<eltBlocks, 256, 0, stream>>>(user_emb, item_emb, acc, hA, hB);

    const long waves      = ((long)N_EDGES + EDGES_PER_WAVE - 1) / EDGES_PER_WAVE;
    const long spmmThr    = waves * 32;
    const int  spmmBlocks = (int)((spmmThr + 255) / 256);

    for (int layer = 0; layer < N_LAYERS; ++layer) {
        lgcn_spmm<<<spmmBlocks, 256, 0, stream>>>(adj_rows, adj_cols, adj_vals, hA, hB);
        lgcn_fuse<<<eltBlocks, 256, 0, stream>>>(acc, hB, hA);
        float* tmp = hA; hA = hB; hB = tmp;    // hB is now zeroed target
    }

    lgcn_dot16<<<N_PAIRS / 128, 256, 0, stream>>>(users, items, acc, out);
}